// GATv2_57174604645032
// MI455X (gfx1250) — compile-verified
//
#include <hip/hip_runtime.h>
#include <math.h>

// ---------------------------------------------------------------------------
// GATv2 (3 layers) for MI455X / gfx1250, wave32.
//   N=10000, E=320000, IN=128, HID=32, HEADS=8, OUT=64, D2=256
// Strategy:
//   * 6 dense GEMMs via V_WMMA_F32_16X16X4_F32 (fp32-exact, edge phase dominates)
//   * CSR-by-destination built once per call (edge_index shared by all layers)
//   * single-pass online segment-softmax aggregation, one wave per (node, head)
// ---------------------------------------------------------------------------

typedef __attribute__((ext_vector_type(2))) float v2f;
typedef __attribute__((ext_vector_type(8))) float v8f;

#define NEG_SLOPE 0.2f

// ---------------------------- CSR construction -----------------------------

__global__ void zero_counts_kernel(int* __restrict__ cnt, int n) {
  int i = blockIdx.x * blockDim.x + threadIdx.x;
  if (i < n) cnt[i] = 0;
}

__global__ void count_edges_kernel(const int* __restrict__ ei, int E,
                                   int* __restrict__ cnt) {
  int e = blockIdx.x * blockDim.x + threadIdx.x;
  if (e < E) atomicAdd(&cnt[ei[E + e]], 1);  // dst = ei[1*E + e]
}

// single-block exclusive scan; counts[] becomes the cursor array afterwards
__global__ void scan_kernel(int* __restrict__ counts, int* __restrict__ off, int n) {
  __shared__ int sh[256];
  int tid = threadIdx.x;
  int run = 0;
  for (int base = 0; base < n; base += 256) {
    int i = base + tid;
    int v = (i < n) ? counts[i] : 0;
    sh[tid] = v;
    __syncthreads();
    for (int o = 1; o < 256; o <<= 1) {
      int t = (tid >= o) ? sh[tid - o] : 0;
      __syncthreads();
      sh[tid] += t;
      __syncthreads();
    }
    int incl = sh[tid];
    int tot  = sh[255];
    if (i < n) {
      int e = run + incl - v;   // exclusive prefix
      off[i]    = e;
      counts[i] = e;            // cursor for fill pass
    }
    run += tot;
    __syncthreads();
  }
  if (tid == 0) off[n] = run;
}

__global__ void fill_csr_kernel(const int* __restrict__ ei, int E,
                                int* __restrict__ cursor,
                                int* __restrict__ csr_src) {
  int e = blockIdx.x * blockDim.x + threadIdx.x;
  if (e < E) {
    int d = ei[E + e];
    int p = atomicAdd(&cursor[d], 1);
    csr_src[p] = ei[e];         // src = ei[0*E + e]
  }
}

// -------------------------- WMMA fp32 GEMM + bias --------------------------
// Y[M,NC] = X[M,K] @ W[K,NC] + bias.  One wave per 16x16 output tile.
// A layout: lane m = lane&15 is the row; v0/v1 hold K = kh, kh+1 (kh = 0 for
// lanes 0-15, 2 for lanes 16-31).  B mirrors A's K packing with lane = column.
__global__ __launch_bounds__(32) void gemm_wmma_kernel(
    const float* __restrict__ X, const float* __restrict__ W,
    const float* __restrict__ bias, float* __restrict__ Y,
    int M, int K, int NC) {
  int lane = threadIdx.x & 31;
  int row0 = blockIdx.x << 4;
  int col0 = blockIdx.y << 4;
  int m  = lane & 15;
  int kh = (lane >> 4) << 1;               // 0 or 2
  int ra = row0 + m;
  if (ra >= M) ra = M - 1;                 // duplicate-load guard (store is masked)

  v8f c = {};
  for (int k = 0; k < K; k += 4) {
    v2f a = *reinterpret_cast<const v2f*>(X + (size_t)ra * K + k + kh);  // 8B aligned
    const float* wp = W + (size_t)(k + kh) * NC + col0 + m;
    v2f b;
    b.x = wp[0];
    b.y = wp[NC];
    c = __builtin_amdgcn_wmma_f32_16x16x4_f32(false, a, false, b,
                                              (short)0, c, false, false);
  }

  float bv = bias[col0 + m];
  int rbase = row0 + ((lane >> 4) << 3);   // rows r / r+8 split across lane halves
  for (int r = 0; r < 8; ++r) {
    int rr = rbase + r;
    if (rr < M) Y[(size_t)rr * NC + col0 + m] = c[r] + bv;
  }
}

// -------------------- online-softmax GATv2 aggregation ---------------------
// One wave per (node, head).  CPL = CH/32 channels per lane.
// ACT: 0 = ELU (layers 1,2), 1 = sigmoid (layer 3 output).
template <int HEADS_T, int CH, int CPL, int ACT>
__global__ __launch_bounds__(HEADS_T * 32) void gat_aggregate_kernel(
    const float* __restrict__ gl, const float* __restrict__ gr,
    const float* __restrict__ att, const float* __restrict__ bias,
    const int* __restrict__ off, const int* __restrict__ csr_src,
    float* __restrict__ out, int n_nodes) {
  int node = blockIdx.x;
  if (node >= n_nodes) return;
  int h    = threadIdx.x >> 5;
  int lane = threadIdx.x & 31;
  const int D = HEADS_T * CH;

  float attv[CPL], grv[CPL], acc[CPL];
#pragma unroll
  for (int cc = 0; cc < CPL; ++cc) {
    int c = h * CH + lane + cc * 32;
    attv[cc] = att[c];
    grv[cc]  = gr[(size_t)node * D + c];
    acc[cc]  = 0.0f;
  }

  float mx = -3.0e38f;   // running max
  float ss = 0.0f;       // running sum of exp(score - mx)

  int e0 = off[node], e1 = off[node + 1];
  for (int i = e0; i < e1; ++i) {
    int s = csr_src[i];
    float val[CPL];
    float t = 0.0f;
#pragma unroll
    for (int cc = 0; cc < CPL; ++cc) {
      float v = gl[(size_t)s * D + h * CH + lane + cc * 32];
      val[cc] = v;
      float z = v + grv[cc];
      z = (z > 0.0f) ? z : NEG_SLOPE * z;          // leaky_relu
      t += attv[cc] * z;
    }
    // full-wave reduction of the score (wave32)
#pragma unroll
    for (int o = 16; o > 0; o >>= 1) t += __shfl_xor(t, o, 32);

    // online softmax update
    float nm    = fmaxf(mx, t);
    float scale = __expf(mx - t > 0.0f ? 0.0f : mx - nm);  // exp(mx-nm), 1 if mx>=t
    float w     = __expf(t - nm);
    ss = ss * scale + w;
#pragma unroll
    for (int cc = 0; cc < CPL; ++cc) acc[cc] = acc[cc] * scale + w * val[cc];
    mx = nm;
  }

  float inv = 1.0f / (ss + 1e-16f);                // empty node: acc==0 -> bias only
#pragma unroll
  for (int cc = 0; cc < CPL; ++cc) {
    int c = h * CH + lane + cc * 32;
    float o = acc[cc] * inv + bias[c];
    if (ACT == 0) o = (o > 0.0f) ? o : (__expf(o) - 1.0f);          // ELU
    else          o = 1.0f / (1.0f + __expf(-o));                    // sigmoid
    out[(size_t)node * D + c] = o;
  }
}

// --------------------------------- launch ----------------------------------

extern "C" void kernel_launch(void* const* d_in, const int* in_sizes, int n_in,
                              void* d_out, int out_size, void* d_ws, size_t ws_size,
                              hipStream_t stream) {
  const float* x    = (const float*)d_in[0];
  const int*   ei   = (const int*)  d_in[1];
  const float* Wl1  = (const float*)d_in[2];
  const float* bl1  = (const float*)d_in[3];
  const float* Wr1  = (const float*)d_in[4];
  const float* br1  = (const float*)d_in[5];
  const float* att1 = (const float*)d_in[6];
  const float* bias1= (const float*)d_in[7];
  const float* Wl2  = (const float*)d_in[8];
  const float* bl2  = (const float*)d_in[9];
  const float* Wr2  = (const float*)d_in[10];
  const float* br2  = (const float*)d_in[11];
  const float* att2 = (const float*)d_in[12];
  const float* bias2= (const float*)d_in[13];
  const float* Wl3  = (const float*)d_in[14];
  const float* bl3  = (const float*)d_in[15];
  const float* Wr3  = (const float*)d_in[16];
  const float* br3  = (const float*)d_in[17];
  const float* att3 = (const float*)d_in[18];
  const float* bias3= (const float*)d_in[19];

  const int IN = 128, D2 = 256, OUT = 64;
  const int N = in_sizes[0] / IN;
  const int E = in_sizes[1] / 2;

  // workspace carve-out (~33 MB)
  char* ws = (char*)d_ws;
  size_t p = 0;
  auto al = [](size_t v) { return (v + 255) & ~(size_t)255; };
  int*   off = (int*)(ws + p);   p = al(p + (size_t)(N + 1) * 4);
  int*   cnt = (int*)(ws + p);   p = al(p + (size_t)N * 4);         // counts -> cursor
  int*   csr = (int*)(ws + p);   p = al(p + (size_t)E * 4);
  float* gl  = (float*)(ws + p); p = al(p + (size_t)N * D2 * 4);
  float* gr  = (float*)(ws + p); p = al(p + (size_t)N * D2 * 4);
  float* hb  = (float*)(ws + p); p = al(p + (size_t)N * D2 * 4);
  (void)ws_size; (void)n_in; (void)out_size;

  // ---- CSR by destination (shared by all three layers) ----
  zero_counts_kernel<<<(N + 255) / 256, 256, 0, stream>>>(cnt, N);
  count_edges_kernel<<<(E + 255) / 256, 256, 0, stream>>>(ei, E, cnt);
  scan_kernel<<<1, 256, 0, stream>>>(cnt, off, N);
  fill_csr_kernel<<<(E + 255) / 256, 256, 0, stream>>>(ei, E, cnt, csr);

  dim3 g16((N + 15) / 16, D2 / 16);
  dim3 g4((N + 15) / 16, OUT / 16);

  // ---- layer 1: IN -> 8x32, ELU ----
  gemm_wmma_kernel<<<g16, 32, 0, stream>>>(x, Wl1, bl1, gl, N, IN, D2);
  gemm_wmma_kernel<<<g16, 32, 0, stream>>>(x, Wr1, br1, gr, N, IN, D2);
  gat_aggregate_kernel<8, 32, 1, 0><<<N, 256, 0, stream>>>(gl, gr, att1, bias1,
                                                           off, csr, hb, N);
  // ---- layer 2: 256 -> 8x32, ELU ----
  gemm_wmma_kernel<<<g16, 32, 0, stream>>>(hb, Wl2, bl2, gl, N, D2, D2);
  gemm_wmma_kernel<<<g16, 32, 0, stream>>>(hb, Wr2, br2, gr, N, D2, D2);
  gat_aggregate_kernel<8, 32, 1, 0><<<N, 256, 0, stream>>>(gl, gr, att2, bias2,
                                                           off, csr, hb, N);
  // ---- layer 3: 256 -> 64 (1 head), sigmoid ----
  gemm_wmma_kernel<<<g4, 32, 0, stream>>>(hb, Wl3, bl3, gl, N, D2, OUT);
  gemm_wmma_kernel<<<g4, 32, 0, stream>>>(hb, Wr3, br3, gr, N, D2, OUT);
  gat_aggregate_kernel<1, 64, 2, 1><<<N, 32, 0, stream>>>(gl, gr, att3, bias3,
                                                          off, csr, (float*)d_out, N);
}